// TGNModel_18210661335214
// MI455X (gfx1250) — compile-verified
//
#include <hip/hip_runtime.h>
#include <math.h>

#define N_NODES  200000
#define N_EVENTS 50000
#define MEMD     128
#define RAW      64
#define TDIM     64
#define EDGE_DIM (RAW + TDIM)            // 128
#define MSG_IN   (2*MEMD + RAW + TDIM)   // 384
#define HEADS    2
#define DH       64

typedef float v2f __attribute__((ext_vector_type(2)));
typedef float v8f __attribute__((ext_vector_type(8)));

// ---------------- workspace layout (element offsets, float-sized slots) ----------------
// weight "P" buffers hold fragment-packed B operands: P[((nt*(K/4))+s)*32 + lane] = float2
static constexpr size_t OFF_WIHP   = 0;                                   // 384x384
static constexpr size_t OFF_WHHP   = OFF_WIHP   + (size_t)384*384;        // 384x128
static constexpr size_t OFF_WQP    = OFF_WHHP   + (size_t)384*128;        // 128x128
static constexpr size_t OFF_WKP    = OFF_WQP    + (size_t)128*128;
static constexpr size_t OFF_WVP    = OFF_WKP    + (size_t)128*128;
static constexpr size_t OFF_WSKIPP = OFF_WVP    + (size_t)128*128;
static constexpr size_t OFF_WLSP   = OFF_WSKIPP + (size_t)128*128;
static constexpr size_t OFF_WLDP   = OFF_WLSP   + (size_t)128*128;
static constexpr size_t OFF_WEP    = OFF_WLDP   + (size_t)128*128;        // 128x192
static constexpr size_t OFF_MEMNEW = OFF_WEP    + (size_t)128*192;        // N*128
static constexpr size_t OFF_LUNEW  = OFF_MEMNEW + (size_t)N_NODES*MEMD;   // N
static constexpr size_t OFF_LASTP  = OFF_LUNEW  + (size_t)N_NODES;        // N (int)
static constexpr size_t OFF_WORK   = OFF_LASTP  + (size_t)N_NODES;        // 2E (int)
static constexpr size_t OFF_CNT    = OFF_WORK   + (size_t)2*N_EVENTS;     // (int)
static constexpr size_t OFF_VE     = OFF_CNT    + 16;                     // E*128
static constexpr size_t OFF_LOGIT  = OFF_VE     + (size_t)N_EVENTS*MEMD;  // E*2
static constexpr size_t OFF_AE     = OFF_LOGIT  + (size_t)N_EVENTS*HEADS; // E*2
static constexpr size_t OFF_MX     = OFF_AE     + (size_t)N_EVENTS*HEADS; // N*2 (int)
static constexpr size_t OFF_DENOM  = OFF_MX     + (size_t)N_NODES*HEADS;  // N*2
static constexpr size_t OFF_OUTA   = OFF_DENOM  + (size_t)N_NODES*HEADS;  // N*128

#define ENC_NEG_INF (-2139095041)   // enc(-inf) = 0xFF800000 ^ 0x7FFFFFFF

__device__ __forceinline__ int enc_f(float f) {
    int i = __float_as_int(f);
    return (i >= 0) ? i : (i ^ 0x7fffffff);   // monotonic int key
}
__device__ __forceinline__ float dec_f(int k) {
    return __int_as_float((k >= 0) ? k : (k ^ 0x7fffffff));
}
__device__ __forceinline__ float sigmoidf_(float x) { return 1.0f / (1.0f + expf(-x)); }

// ---- fp32 WMMA 16x16 tile accumulate over full K ----
// A (16xK) in LDS, padded lda (even) -> 8B v2f fragment loads
// B packed fragment-native: one global_load_b64 per WMMA, 256B linear stream per wave
// C frag (16x16 f32): vgpr r -> row = r + 8*(lane>>4), col = lane&15
__device__ __forceinline__ v8f wmma_acc_pk(const float* __restrict__ A, int lda,
                                           const v2f* __restrict__ Bp, int kTot,
                                           int nt, v8f c)
{
    const int lane = threadIdx.x & 31;
    const int rc   = lane & 15;          // A row / C col
    const int koff = (lane >> 4) << 1;   // 0 or 2
    const int kq   = kTot >> 2;
    const v2f* bq  = Bp + (size_t)nt * kq * 32 + lane;
    for (int s = 0; s < kq; ++s) {
        v2f a = *(const v2f*)(A + rc * lda + (s << 2) + koff);
        v2f b = bq[(size_t)s * 32];
        c = __builtin_amdgcn_wmma_f32_16x16x4_f32(false, a, false, b, (short)0, c, false, false);
    }
    return c;
}

// ---------------------------------------------------------------------------------------
__global__ void k_init(int* last_pos, int* counter) {
    int idx = blockIdx.x * blockDim.x + threadIdx.x;
    if (idx < N_NODES) last_pos[idx] = -1;
    if (idx == 0) counter[0] = 0;
}

// pack W (NxK row-major) into WMMA B-fragment order:
// P[((nt*(K/4))+s)*32 + lane] = { W[n][k], W[n][k+1] }, n=nt*16+(lane&15), k=4s+2*(lane>>4)
__global__ void k_pack(const float* __restrict__ W, v2f* __restrict__ P, int N, int K) {
    int idx = blockIdx.x * blockDim.x + threadIdx.x;
    int total = (N >> 4) * (K >> 2) * 32;
    if (idx >= total) return;
    int lane = idx & 31;
    int rem  = idx >> 5;
    int kq   = K >> 2;
    int s    = rem % kq;
    int nt   = rem / kq;
    int n = nt * 16 + (lane & 15);
    int k = (s << 2) + ((lane >> 4) << 1);
    v2f v;
    v.x = W[(size_t)n * K + k];
    v.y = W[(size_t)n * K + k + 1];
    P[idx] = v;
}

__global__ void k_lastpos(const int* __restrict__ src, const int* __restrict__ dst, int* last_pos) {
    int idx = blockIdx.x * blockDim.x + threadIdx.x;
    if (idx >= 2 * N_EVENTS) return;
    int id = (idx < N_EVENTS) ? src[idx] : dst[idx - N_EVENTS];
    atomicMax(&last_pos[id], idx);
}

__global__ void k_compact(const int* __restrict__ src, const int* __restrict__ dst,
                          const int* __restrict__ last_pos, int* worklist, int* counter) {
    int idx = blockIdx.x * blockDim.x + threadIdx.x;
    if (idx >= 2 * N_EVENTS) return;
    int id = (idx < N_EVENTS) ? src[idx] : dst[idx - N_EVENTS];
    if (last_pos[id] == idx) {
        int w = atomicAdd(counter, 1);
        worklist[w] = idx;
    }
}

// -------- GRU on compacted last-messages; scatters memory_new / last_update_new --------
__global__ __launch_bounds__(128) void k_gru(
    const float* __restrict__ memory, const float* __restrict__ last_update,
    const float* __restrict__ t, const float* __restrict__ msg,
    const int* __restrict__ src, const int* __restrict__ dst,
    const float* __restrict__ time_w, const float* __restrict__ time_b,
    const float* __restrict__ bih, const float* __restrict__ bhh,
    const v2f* __restrict__ WihP, const v2f* __restrict__ WhhP,
    const int* __restrict__ worklist, const int* __restrict__ counter,
    float* __restrict__ mem_new, float* __restrict__ lu_new)
{
    __shared__ float Msg[16][MSG_IN + 4];   // stride 388 (even): 8B-aligned frags, no bank camping
    __shared__ int   aid[16], bid[16], eid[16], vld[16];
    __shared__ float tv[16], dtv[16];

    const int tid = threadIdx.x;
    const int count = counter[0];
    const int base = blockIdx.x * 16;
    if (base >= count) return;

    if (tid < 16) {
        int j = base + tid;
        int a = 0, b = 0, e = 0, ok = 0;
        float tt = 0.f, dt = 0.f;
        if (j < count) {
            int m = worklist[j];
            e = (m < N_EVENTS) ? m : m - N_EVENTS;
            a = (m < N_EVENTS) ? src[e] : dst[e];
            b = (m < N_EVENTS) ? dst[e] : src[e];
            ok = 1;
            tt = t[e];
            dt = tt - last_update[a];
        }
        aid[tid] = a; bid[tid] = b; eid[tid] = e; vld[tid] = ok; tv[tid] = tt; dtv[tid] = dt;
    }
    __syncthreads();

    // build message tile: [mem[a] | mem[b] | raw msg | time_enc(dt)]
    for (int idx = tid; idx < 16 * MSG_IN; idx += 128) {
        int row = idx / MSG_IN, col = idx - row * MSG_IN;
        float vv;
        if (col < MEMD)                vv = memory[(size_t)aid[row] * MEMD + col];
        else if (col < 2 * MEMD)       vv = memory[(size_t)bid[row] * MEMD + (col - MEMD)];
        else if (col < 2 * MEMD + RAW) vv = msg[(size_t)eid[row] * RAW + (col - 2 * MEMD)];
        else { int d = col - (2 * MEMD + RAW); vv = cosf(dtv[row] * time_w[d] + time_b[d]); }
        Msg[row][col] = vv;
    }
    __syncthreads();

    const int wave  = tid >> 5;
    const int lane  = tid & 31;
    const int col16 = lane & 15;
    const int mbase = (lane >> 4) << 3;

    // wave w owns N-tiles {cb, cb+8, cb+16} for cb in {w, w+4}  -> gate slices stay in-wave
    v8f gi[2][3], gh[2][3];
    #pragma unroll
    for (int cbi = 0; cbi < 2; ++cbi) {
        int cb = wave + 4 * cbi;
        #pragma unroll
        for (int g = 0; g < 3; ++g) {
            int nt = cb + 8 * g;
            v8f ci = {};
            ci = wmma_acc_pk(&Msg[0][0], MSG_IN + 4, WihP, MSG_IN, nt, ci);
            gi[cbi][g] = ci;
            v8f ch = {};
            ch = wmma_acc_pk(&Msg[0][0], MSG_IN + 4, WhhP, MEMD, nt, ch);  // h = first 128 cols
            gh[cbi][g] = ch;
        }
    }

    // gates + scatter (divergence only after all WMMAs)
    #pragma unroll
    for (int cbi = 0; cbi < 2; ++cbi) {
        int cb = wave + 4 * cbi;
        int d = cb * 16 + col16;   // [0,128)
        #pragma unroll
        for (int r = 0; r < 8; ++r) {
            int row = mbase + r;
            float rg = sigmoidf_(gi[cbi][0][r] + bih[d]            + gh[cbi][0][r] + bhh[d]);
            float zg = sigmoidf_(gi[cbi][1][r] + bih[MEMD + d]     + gh[cbi][1][r] + bhh[MEMD + d]);
            float ng = tanhf(    gi[cbi][2][r] + bih[2 * MEMD + d] + rg * (gh[cbi][2][r] + bhh[2 * MEMD + d]));
            float h  = Msg[row][d];
            float hv = (1.0f - zg) * ng + zg * h;
            if (vld[row]) mem_new[(size_t)aid[row] * MEMD + d] = hv;
        }
    }
    if (tid < 16 && vld[tid]) {
        int a = aid[tid];
        lu_new[a] = fmaxf(last_update[a], tv[tid]);
    }
}

// -------- zero attention accumulators at all touched rows; init mx/denom at dst --------
__global__ void k_att_init(const int* __restrict__ src, const int* __restrict__ dst,
                           float* out_attn, int* mx_enc, float* denom) {
    size_t idx = (size_t)blockIdx.x * blockDim.x + threadIdx.x;
    if (idx >= (size_t)N_EVENTS * MEMD) return;
    int e = (int)(idx >> 7), d = (int)(idx & 127);
    out_attn[(size_t)src[e] * MEMD + d] = 0.f;
    out_attn[(size_t)dst[e] * MEMD + d] = 0.f;
    if (d < HEADS) {
        mx_enc[(size_t)dst[e] * HEADS + d] = ENC_NEG_INF;
        denom[(size_t)dst[e] * HEADS + d] = 0.f;
    }
}

// -------- per-edge e_proj / K / V / Q GEMMs + logits + segment max --------
__global__ __launch_bounds__(128) void k_edge(
    const float* __restrict__ mem_new, const float* __restrict__ lu_new,
    const float* __restrict__ t, const float* __restrict__ msg,
    const int* __restrict__ src, const int* __restrict__ dst,
    const float* __restrict__ time_w, const float* __restrict__ time_b,
    const v2f* __restrict__ WeP, const v2f* __restrict__ WkP,
    const v2f* __restrict__ WvP, const v2f* __restrict__ WqP,
    const float* __restrict__ bk, const float* __restrict__ bv, const float* __restrict__ bq,
    float* __restrict__ v_ws, float* __restrict__ logit_ws, int* __restrict__ mx_enc)
{
    __shared__ float EA[16][EDGE_DIM + 4];
    __shared__ float MS[16][MEMD + 4];
    __shared__ float MD[16][MEMD + 4];
    __shared__ float KT[16][MEMD + 4];
    __shared__ float QT[16][MEMD + 4];

    const int tid = threadIdx.x;
    const int e0 = blockIdx.x * 16;     // E = 50000 = 3125 * 16, no partial tiles

    for (int idx = tid; idx < 16 * EDGE_DIM; idx += 128) {
        int row = idx / EDGE_DIM, col = idx - row * EDGE_DIM;
        int e = e0 + row;
        float vv;
        if (col < TDIM) {
            float rel = lu_new[src[e]] - t[e];
            vv = cosf(rel * time_w[col] + time_b[col]);
        } else vv = msg[(size_t)e * RAW + (col - TDIM)];
        EA[row][col] = vv;
    }
    for (int idx = tid; idx < 16 * MEMD; idx += 128) {
        int row = idx >> 7, col = idx & 127;
        MS[row][col] = mem_new[(size_t)src[e0 + row] * MEMD + col];
        MD[row][col] = mem_new[(size_t)dst[e0 + row] * MEMD + col];
    }
    __syncthreads();

    const int wave  = tid >> 5;
    const int lane  = tid & 31;
    const int col16 = lane & 15;
    const int mbase = (lane >> 4) << 3;

    #pragma unroll
    for (int i = 0; i < 2; ++i) {
        int nt = wave + 4 * i;
        v8f ep = {}; ep = wmma_acc_pk(&EA[0][0], EDGE_DIM + 4, WeP, EDGE_DIM, nt, ep);
        v8f kc = {}; kc = wmma_acc_pk(&MS[0][0], MEMD + 4,     WkP, MEMD,     nt, kc);
        v8f vc = {}; vc = wmma_acc_pk(&MS[0][0], MEMD + 4,     WvP, MEMD,     nt, vc);
        v8f qc = {}; qc = wmma_acc_pk(&MD[0][0], MEMD + 4,     WqP, MEMD,     nt, qc);
        int n = nt * 16 + col16;
        #pragma unroll
        for (int r = 0; r < 8; ++r) {
            int row = mbase + r;
            KT[row][n] = kc[r] + ep[r] + bk[n];
            QT[row][n] = qc[r] + bq[n];
            v_ws[(size_t)(e0 + row) * MEMD + n] = vc[r] + ep[r] + bv[n];
        }
    }
    __syncthreads();

    if (tid < 32) {
        int i = tid >> 1, h = tid & 1;
        float s = 0.f;
        for (int d = 0; d < DH; ++d) s += QT[i][h * DH + d] * KT[i][h * DH + d];
        s *= 0.125f;   // 1/sqrt(64)
        int e = e0 + i;
        logit_ws[(size_t)e * HEADS + h] = s;
        atomicMax(&mx_enc[(size_t)dst[e] * HEADS + h], enc_f(s));
    }
}

__global__ void k_softmax_a(const int* __restrict__ dst, const float* __restrict__ logit_ws,
                            const int* __restrict__ mx_enc, float* a_ws, float* denom) {
    int idx = blockIdx.x * blockDim.x + threadIdx.x;
    if (idx >= N_EVENTS * HEADS) return;
    int e = idx >> 1, h = idx & 1;
    float m = dec_f(mx_enc[(size_t)dst[e] * HEADS + h]);
    float a = expf(logit_ws[idx] - m);
    a_ws[idx] = a;
    atomicAdd(&denom[(size_t)dst[e] * HEADS + h], a);
}

__global__ void k_scatter_v(const int* __restrict__ dst, const float* __restrict__ a_ws,
                            const float* __restrict__ denom, const float* __restrict__ v_ws,
                            float* out_attn) {
    size_t idx = (size_t)blockIdx.x * blockDim.x + threadIdx.x;
    if (idx >= (size_t)N_EVENTS * MEMD) return;
    int e = (int)(idx >> 7), d = (int)(idx & 127), h = d >> 6;
    float attn = a_ws[(size_t)e * HEADS + h] / denom[(size_t)dst[e] * HEADS + h];
    atomicAdd(&out_attn[(size_t)dst[e] * MEMD + d], attn * v_ws[idx]);
}

// -------- skip GEMM + link predictor, fused per 16-edge tile --------
__global__ __launch_bounds__(128) void k_link(
    const float* __restrict__ mem_new, const float* __restrict__ out_attn,
    const int* __restrict__ src, const int* __restrict__ dst,
    const v2f* __restrict__ WskipP, const float* __restrict__ bskip,
    const v2f* __restrict__ WlsP, const float* __restrict__ bls,
    const v2f* __restrict__ WldP, const float* __restrict__ bld,
    const float* __restrict__ Wlf, const float* __restrict__ blf,
    float* __restrict__ out)
{
    __shared__ float MS[16][MEMD + 4];
    __shared__ float MD[16][MEMD + 4];
    __shared__ float ZS[16][MEMD + 4];
    __shared__ float ZD[16][MEMD + 4];
    __shared__ float HH[16][MEMD + 4];

    const int tid = threadIdx.x;
    const int e0 = blockIdx.x * 16;

    for (int idx = tid; idx < 16 * MEMD; idx += 128) {
        int row = idx >> 7, col = idx & 127;
        MS[row][col] = mem_new[(size_t)src[e0 + row] * MEMD + col];
        MD[row][col] = mem_new[(size_t)dst[e0 + row] * MEMD + col];
    }
    __syncthreads();

    const int wave  = tid >> 5;
    const int lane  = tid & 31;
    const int col16 = lane & 15;
    const int mbase = (lane >> 4) << 3;

    #pragma unroll
    for (int i = 0; i < 2; ++i) {
        int nt = wave + 4 * i;
        v8f zs = {}; zs = wmma_acc_pk(&MS[0][0], MEMD + 4, WskipP, MEMD, nt, zs);
        v8f zd = {}; zd = wmma_acc_pk(&MD[0][0], MEMD + 4, WskipP, MEMD, nt, zd);
        int n = nt * 16 + col16;
        #pragma unroll
        for (int r = 0; r < 8; ++r) {
            int row = mbase + r;
            ZS[row][n] = zs[r] + out_attn[(size_t)src[e0 + row] * MEMD + n] + bskip[n];
            ZD[row][n] = zd[r] + out_attn[(size_t)dst[e0 + row] * MEMD + n] + bskip[n];
        }
    }
    __syncthreads();

    #pragma unroll
    for (int i = 0; i < 2; ++i) {
        int nt = wave + 4 * i;
        v8f hc = {};
        hc = wmma_acc_pk(&ZS[0][0], MEMD + 4, WlsP, MEMD, nt, hc);
        hc = wmma_acc_pk(&ZD[0][0], MEMD + 4, WldP, MEMD, nt, hc);
        int n = nt * 16 + col16;
        #pragma unroll
        for (int r = 0; r < 8; ++r) {
            int row = mbase + r;
            HH[row][n] = fmaxf(hc[r] + bls[n] + bld[n], 0.0f);
        }
    }
    __syncthreads();

    if (tid < 16) {
        float s = blf[0];
        for (int d = 0; d < MEMD; ++d) s += HH[tid][d] * Wlf[d];
        out[e0 + tid] = s;
    }
}

// ---------------------------------------------------------------------------------------
extern "C" void kernel_launch(void* const* d_in, const int* in_sizes, int n_in,
                              void* d_out, int out_size, void* d_ws, size_t ws_size,
                              hipStream_t stream) {
    const float* memory      = (const float*)d_in[0];
    const float* last_update = (const float*)d_in[1];
    const float* t           = (const float*)d_in[2];
    const float* msg         = (const float*)d_in[3];
    const int*   src         = (const int*)d_in[4];
    const int*   dst         = (const int*)d_in[5];
    const float* time_w      = (const float*)d_in[6];
    const float* time_b      = (const float*)d_in[7];
    const float* gru_Wih     = (const float*)d_in[8];
    const float* gru_bih     = (const float*)d_in[9];
    const float* gru_Whh     = (const float*)d_in[10];
    const float* gru_bhh     = (const float*)d_in[11];
    const float* Wq          = (const float*)d_in[12];
    const float* bq          = (const float*)d_in[13];
    const float* Wk          = (const float*)d_in[14];
    const float* bk          = (const float*)d_in[15];
    const float* Wv          = (const float*)d_in[16];
    const float* bv          = (const float*)d_in[17];
    const float* We          = (const float*)d_in[18];
    const float* Wskip       = (const float*)d_in[19];
    const float* bskip       = (const float*)d_in[20];
    const float* Wls         = (const float*)d_in[21];
    const float* bls         = (const float*)d_in[22];
    const float* Wld         = (const float*)d_in[23];
    const float* bld         = (const float*)d_in[24];
    const float* Wlf         = (const float*)d_in[25];
    const float* blf         = (const float*)d_in[26];

    float* ws      = (float*)d_ws;
    v2f*   WihP    = (v2f*)(ws + OFF_WIHP);
    v2f*   WhhP    = (v2f*)(ws + OFF_WHHP);
    v2f*   WqP     = (v2f*)(ws + OFF_WQP);
    v2f*   WkP     = (v2f*)(ws + OFF_WKP);
    v2f*   WvP     = (v2f*)(ws + OFF_WVP);
    v2f*   WskipP  = (v2f*)(ws + OFF_WSKIPP);
    v2f*   WlsP    = (v2f*)(ws + OFF_WLSP);
    v2f*   WldP    = (v2f*)(ws + OFF_WLDP);
    v2f*   WeP     = (v2f*)(ws + OFF_WEP);
    float* mem_new = ws + OFF_MEMNEW;
    float* lu_new  = ws + OFF_LUNEW;
    int*   lastp   = (int*)(ws + OFF_LASTP);
    int*   work    = (int*)(ws + OFF_WORK);
    int*   counter = (int*)(ws + OFF_CNT);
    float* v_ws    = ws + OFF_VE;
    float* logit   = ws + OFF_LOGIT;
    float* a_ws    = ws + OFF_AE;
    int*   mx_enc  = (int*)(ws + OFF_MX);
    float* denom   = ws + OFF_DENOM;
    float* out_a   = ws + OFF_OUTA;
    float* outp    = (float*)d_out;

    // 1) init + weight fragment-packing (B operands become linear b64 streams)
    k_init<<<(N_NODES + 255) / 256, 256, 0, stream>>>(lastp, counter);
    k_pack<<<(384 * 384 / 2 + 255) / 256, 256, 0, stream>>>(gru_Wih, WihP, 384, 384);
    k_pack<<<(384 * 128 / 2 + 255) / 256, 256, 0, stream>>>(gru_Whh, WhhP, 384, 128);
    k_pack<<<(128 * 128 / 2 + 255) / 256, 256, 0, stream>>>(Wq, WqP, 128, 128);
    k_pack<<<(128 * 128 / 2 + 255) / 256, 256, 0, stream>>>(Wk, WkP, 128, 128);
    k_pack<<<(128 * 128 / 2 + 255) / 256, 256, 0, stream>>>(Wv, WvP, 128, 128);
    k_pack<<<(128 * 128 / 2 + 255) / 256, 256, 0, stream>>>(Wskip, WskipP, 128, 128);
    k_pack<<<(128 * 128 / 2 + 255) / 256, 256, 0, stream>>>(Wls, WlsP, 128, 128);
    k_pack<<<(128 * 128 / 2 + 255) / 256, 256, 0, stream>>>(Wld, WldP, 128, 128);
    k_pack<<<(128 * 192 / 2 + 255) / 256, 256, 0, stream>>>(We, WeP, 128, 192);

    // 2) last-message selection + compaction
    k_lastpos<<<(2 * N_EVENTS + 255) / 256, 256, 0, stream>>>(src, dst, lastp);
    k_compact<<<(2 * N_EVENTS + 255) / 256, 256, 0, stream>>>(src, dst, lastp, work, counter);

    // 3) GRU memory update on surviving messages only
    k_gru<<<(2 * N_EVENTS) / 16, 128, 0, stream>>>(
        memory, last_update, t, msg, src, dst, time_w, time_b,
        gru_bih, gru_bhh, WihP, WhhP, work, counter, mem_new, lu_new);

    // 4) attention: init, per-edge GEMMs + logits + segment-max, softmax, scatter
    k_att_init<<<(N_EVENTS * MEMD + 255) / 256, 256, 0, stream>>>(src, dst, out_a, mx_enc, denom);
    k_edge<<<N_EVENTS / 16, 128, 0, stream>>>(
        mem_new, lu_new, t, msg, src, dst, time_w, time_b,
        WeP, WkP, WvP, WqP, bk, bv, bq, v_ws, logit, mx_enc);
    k_softmax_a<<<(N_EVENTS * HEADS + 255) / 256, 256, 0, stream>>>(dst, logit, mx_enc, a_ws, denom);
    k_scatter_v<<<(N_EVENTS * MEMD + 255) / 256, 256, 0, stream>>>(dst, a_ws, denom, v_ws, out_a);

    // 5) fused skip GEMM + link predictor
    k_link<<<N_EVENTS / 16, 128, 0, stream>>>(
        mem_new, out_a, src, dst, WskipP, bskip, WlsP, bls, WldP, bld, Wlf, blf, outp);
}